// GraphResNetBlockGAT_13752485282057
// MI455X (gfx1250) — compile-verified
//
#include <hip/hip_runtime.h>
#include <math.h>

#define HN 4
#define CN 16
#define DN 64
#define BN_EPS 1e-5f

typedef __attribute__((ext_vector_type(2))) float v2f;
typedef __attribute__((ext_vector_type(8))) float v8f;

// ---------- monotone float<->uint mapping for atomic segment-max ----------
__device__ __forceinline__ unsigned fmap(float f) {
  unsigned u = __float_as_uint(f);
  return (u & 0x80000000u) ? ~u : (u | 0x80000000u);
}
__device__ __forceinline__ float funmap(unsigned m) {
  unsigned u = (m & 0x80000000u) ? (m ^ 0x80000000u) : ~m;
  return __uint_as_float(u);
}

// ---------- sum reduction (edge_attr mean) ----------
__global__ void k_reduce_sum(const float* __restrict__ v, float* __restrict__ out, int n) {
  __shared__ float sh[256];
  float s = 0.f;
  for (int i = blockIdx.x * blockDim.x + threadIdx.x; i < n; i += gridDim.x * blockDim.x)
    s += v[i];
  sh[threadIdx.x] = s;
  __syncthreads();
  for (int off = 128; off > 0; off >>= 1) {
    if ((int)threadIdx.x < off) sh[threadIdx.x] += sh[threadIdx.x + off];
    __syncthreads();
  }
  if (threadIdx.x == 0) atomicAdd(out, sh[0]);
}

// small[] layout (floats):
// [0] ea_sum  [1] ea_mean  [4..7] we_dot[4]
// [8..71] bn_sum  [72..135] bn_sumsq  [136..199] bn_mean  [200..263] bn_inv_std
__global__ void k_prep(float* __restrict__ small, const float* __restrict__ We,
                       const float* __restrict__ ae, float invE) {
  int t = threadIdx.x;
  if (t == 0) small[1] = small[0] * invE;           // ea_mean
  if (t < HN) {
    float s = 0.f;
    #pragma unroll
    for (int c = 0; c < CN; ++c) s += We[t * CN + c] * ae[t * CN + c];
    small[4 + t] = s;                                // we_dot[h]
  }
}

// ---------- H = X (N x 64) @ W^T (64 x 64) via fp32 WMMA 16x16x4 ----------
__global__ __launch_bounds__(128) void k_gemm_wmma(const float* __restrict__ X,
                                                   const float* __restrict__ W,
                                                   float* __restrict__ Hout, int nrows) {
  const int wave = threadIdx.x >> 5;   // 0..3 -> 16-col block
  const int lane = threadIdx.x & 31;
  const int half = lane >> 4;          // 0/1
  const int l16  = lane & 15;
  const int row0 = blockIdx.x * 16;
  const int col0 = wave * 16;

  v8f acc = {};
  int arow = row0 + l16;
  if (arow >= nrows) arow = nrows - 1;   // clamp (stores guarded below)
  const int bcol = col0 + l16;

  #pragma unroll
  for (int kt = 0; kt < 16; ++kt) {
    const int kb = kt * 4 + half * 2;
    v2f a, b;
    a.x = X[(size_t)arow * DN + kb];
    a.y = X[(size_t)arow * DN + kb + 1];
    b.x = W[(size_t)bcol * DN + kb];     // B[k,n] = W[n,k]  (h = x @ W^T)
    b.y = W[(size_t)bcol * DN + kb + 1];
    acc = __builtin_amdgcn_wmma_f32_16x16x4_f32(false, a, false, b,
                                                (short)0, acc, false, false);
  }
  #pragma unroll
  for (int r = 0; r < 8; ++r) {
    const int row = row0 + (half ? r + 8 : r);
    if (row < nrows) Hout[(size_t)row * DN + col0 + l16] = acc[r];
  }
}

// ---------- per-(node,head) attention coefficients ----------
__global__ void k_attn_coef(const float* __restrict__ Hb, const float* __restrict__ as_,
                            const float* __restrict__ ad_, float* __restrict__ asrc,
                            float* __restrict__ adst, int N) {
  const int total = N * HN;
  for (int t = blockIdx.x * blockDim.x + threadIdx.x; t < total; t += gridDim.x * blockDim.x) {
    const int n = t >> 2, hh = t & 3;
    const float4* hp = (const float4*)(Hb + (size_t)n * DN + hh * CN);
    const float4* sp = (const float4*)(as_ + hh * CN);
    const float4* dp = (const float4*)(ad_ + hh * CN);
    float s = 0.f, d = 0.f;
    #pragma unroll
    for (int j = 0; j < 4; ++j) {
      float4 hv = hp[j], sv = sp[j], dv = dp[j];
      s += hv.x * sv.x + hv.y * sv.y + hv.z * sv.z + hv.w * sv.w;
      d += hv.x * dv.x + hv.y * dv.y + hv.z * dv.z + hv.w * dv.w;
    }
    asrc[t] = s;
    adst[t] = d;
  }
}

// ---------- edge pass A: raw logits (leaky-relu) + segment max ----------
__global__ void k_edge_alpha(const int* __restrict__ esrc, const int* __restrict__ edst,
                             const float* __restrict__ ea, const float* __restrict__ small,
                             const float* __restrict__ asrc, const float* __restrict__ adst,
                             float* __restrict__ etmp, unsigned* __restrict__ amax,
                             int E, int EP) {
  for (int e = blockIdx.x * blockDim.x + threadIdx.x; e < EP; e += gridDim.x * blockDim.x) {
    int s_, d_;
    float eav;
    if (e < E) { s_ = esrc[e]; d_ = edst[e]; eav = ea[e]; }
    else       { s_ = d_ = e - E;            eav = small[1]; }
    const float4 av = *(const float4*)(asrc + (size_t)s_ * HN);
    const float4 dv = *(const float4*)(adst + (size_t)d_ * HN);
    const float4 wd = *(const float4*)(small + 4);
    float al[4];
    al[0] = av.x + dv.x + eav * wd.x;
    al[1] = av.y + dv.y + eav * wd.y;
    al[2] = av.z + dv.z + eav * wd.z;
    al[3] = av.w + dv.w + eav * wd.w;
    #pragma unroll
    for (int hh = 0; hh < HN; ++hh) {
      al[hh] = (al[hh] > 0.f) ? al[hh] : 0.2f * al[hh];
      atomicMax(&amax[(size_t)d_ * HN + hh], fmap(al[hh]));
    }
    float4 st; st.x = al[0]; st.y = al[1]; st.z = al[2]; st.w = al[3];
    *(float4*)(etmp + (size_t)e * HN) = st;
  }
}

// ---------- edge pass B: exp(alpha - max) + segment sum ----------
__global__ void k_edge_exp(const int* __restrict__ edst, float* __restrict__ etmp,
                           const unsigned* __restrict__ amax, float* __restrict__ den,
                           int E, int EP) {
  for (int e = blockIdx.x * blockDim.x + threadIdx.x; e < EP; e += gridDim.x * blockDim.x) {
    const int d_ = (e < E) ? edst[e] : (e - E);
    float4 al = *(const float4*)(etmp + (size_t)e * HN);
    float ex[4];
    ex[0] = __expf(al.x - funmap(amax[(size_t)d_ * HN + 0]));
    ex[1] = __expf(al.y - funmap(amax[(size_t)d_ * HN + 1]));
    ex[2] = __expf(al.z - funmap(amax[(size_t)d_ * HN + 2]));
    ex[3] = __expf(al.w - funmap(amax[(size_t)d_ * HN + 3]));
    float4 st; st.x = ex[0]; st.y = ex[1]; st.z = ex[2]; st.w = ex[3];
    *(float4*)(etmp + (size_t)e * HN) = st;
    #pragma unroll
    for (int hh = 0; hh < HN; ++hh)
      atomicAdd(&den[(size_t)d_ * HN + hh], ex[hh]);
  }
}

// ---------- edge pass C: weighted aggregation (per edge,head) ----------
__global__ void k_edge_aggr(const int* __restrict__ esrc, const int* __restrict__ edst,
                            const float* __restrict__ etmp, const float* __restrict__ den,
                            const float* __restrict__ Hb, float* __restrict__ gout,
                            int E, int EP) {
  const int total = EP * HN;
  for (int t = blockIdx.x * blockDim.x + threadIdx.x; t < total; t += gridDim.x * blockDim.x) {
    const int e = t >> 2, hh = t & 3;
    int s_, d_;
    if (e < E) { s_ = esrc[e]; d_ = edst[e]; }
    else       { s_ = d_ = e - E; }
    const float w = etmp[(size_t)e * HN + hh] / den[(size_t)d_ * HN + hh];
    const float4* hp = (const float4*)(Hb + (size_t)s_ * DN + hh * CN);
    float* gp = gout + (size_t)d_ * DN + hh * CN;
    #pragma unroll
    for (int j = 0; j < 4; ++j) {
      float4 hv = hp[j];
      atomicAdd(gp + j * 4 + 0, w * hv.x);
      atomicAdd(gp + j * 4 + 1, w * hv.y);
      atomicAdd(gp + j * 4 + 2, w * hv.z);
      atomicAdd(gp + j * 4 + 3, w * hv.w);
    }
  }
}

// ---------- BatchNorm statistics (block-staged) ----------
__global__ __launch_bounds__(256) void k_bn_stats(const float* __restrict__ g,
                                                  const float* __restrict__ b,
                                                  float* __restrict__ small, int N) {
  __shared__ float ss[4][DN];
  __shared__ float ss2[4][DN];
  const int d = threadIdx.x & 63;
  const int r = threadIdx.x >> 6;      // 0..3
  const float bd = b[d];
  float s = 0.f, s2 = 0.f;
  for (int n = blockIdx.x * 4 + r; n < N; n += gridDim.x * 4) {
    const float v = g[(size_t)n * DN + d] + bd;
    s += v; s2 += v * v;
  }
  ss[r][d] = s; ss2[r][d] = s2;
  __syncthreads();
  if (r == 0) {
    s  = ss[0][d]  + ss[1][d]  + ss[2][d]  + ss[3][d];
    s2 = ss2[0][d] + ss2[1][d] + ss2[2][d] + ss2[3][d];
    atomicAdd(&small[8 + d], s);
    atomicAdd(&small[72 + d], s2);
  }
}

__global__ void k_bn_final(float* __restrict__ small, float invN) {
  const int d = threadIdx.x;
  if (d < DN) {
    const float mean = small[8 + d] * invN;
    const float var  = small[72 + d] * invN - mean * mean;
    small[136 + d] = mean;
    small[200 + d] = rsqrtf(var + BN_EPS);
  }
}

// ---------- fused BN + (residual) + ReLU ----------
__global__ void k_apply(const float* __restrict__ g, const float* __restrict__ b,
                        const float* __restrict__ small, const float* __restrict__ gamma,
                        const float* __restrict__ beta, const float* __restrict__ resid,
                        float* __restrict__ out, int N) {
  const int total = N * DN;
  for (int i = blockIdx.x * blockDim.x + threadIdx.x; i < total; i += gridDim.x * blockDim.x) {
    const int d = i & 63;
    float v = (g[i] + b[d] - small[136 + d]) * small[200 + d] * gamma[d] + beta[d];
    if (resid) v += resid[i];
    out[i] = v > 0.f ? v : 0.f;
  }
}

extern "C" void kernel_launch(void* const* d_in, const int* in_sizes, int n_in,
                              void* d_out, int out_size, void* d_ws, size_t ws_size,
                              hipStream_t stream) {
  const float* x     = (const float*)d_in[0];
  const int*   ei    = (const int*)d_in[1];
  const float* ea    = (const float*)d_in[2];
  // d_in[3] = batch (unused)
  const float* W1    = (const float*)d_in[4];
  const float* as1   = (const float*)d_in[5];
  const float* ad1   = (const float*)d_in[6];
  const float* ae1   = (const float*)d_in[7];
  const float* We1   = (const float*)d_in[8];
  const float* b1    = (const float*)d_in[9];
  const float* W2    = (const float*)d_in[10];
  const float* as2   = (const float*)d_in[11];
  const float* ad2   = (const float*)d_in[12];
  const float* ae2   = (const float*)d_in[13];
  const float* We2   = (const float*)d_in[14];
  const float* b2    = (const float*)d_in[15];
  const float* gamma = (const float*)d_in[16];
  const float* beta  = (const float*)d_in[17];
  float* out = (float*)d_out;

  const int N  = in_sizes[0] / DN;   // 100000
  const int E  = in_sizes[2];        // 1600000
  const int EP = E + N;
  const int* esrc = ei;
  const int* edst = ei + E;

  // ---- workspace layout ----
  char* ws = (char*)d_ws;
  float* hbuf = (float*)ws;                 ws += (size_t)N * DN * sizeof(float);
  float* xmid = (float*)ws;                 ws += (size_t)N * DN * sizeof(float);
  float* gout = (float*)ws;                 ws += (size_t)N * DN * sizeof(float);
  float* etmp = (float*)ws;                 ws += (size_t)EP * HN * sizeof(float);
  float* asrc = (float*)ws;                 ws += (size_t)N * HN * sizeof(float);
  float* adst = (float*)ws;                 ws += (size_t)N * HN * sizeof(float);
  unsigned* amax = (unsigned*)ws;           ws += (size_t)N * HN * sizeof(unsigned);
  float* den  = (float*)ws;                 ws += (size_t)N * HN * sizeof(float);
  float* small = (float*)ws;                /* 264 floats */

  const int TB = 256;
  const int gbEdge = (EP + TB - 1) / TB;
  const int gbEH   = (EP * HN + TB - 1) / TB;
  const int gbNH   = (N * HN + TB - 1) / TB;
  const int gbND   = (N * DN + TB - 1) / TB;
  const int gbGemm = (N + 15) / 16;

  // ---- one-time: edge_attr mean accumulator ----
  hipMemsetAsync(small, 0, 264 * sizeof(float), stream);
  k_reduce_sum<<<512, TB, 0, stream>>>(ea, small, E);

  struct LayerP {
    const float *xin, *W, *as_, *ad_, *ae_, *We_, *b_;
    const float* resid;
    float* xout;
  };
  LayerP layers[2] = {
    { x,    W1, as1, ad1, ae1, We1, b1, nullptr, xmid },
    { xmid, W2, as2, ad2, ae2, We2, b2, x,       out  },
  };

  for (int L = 0; L < 2; ++L) {
    const LayerP& p = layers[L];
    // reset per-layer accumulators (keep small[0..7] = ea_sum region)
    hipMemsetAsync(small + 8, 0, 256 * sizeof(float), stream);
    hipMemsetAsync(amax, 0, (size_t)N * HN * sizeof(unsigned), stream);  // fmap-min
    hipMemsetAsync(den,  0, (size_t)N * HN * sizeof(float), stream);
    hipMemsetAsync(gout, 0, (size_t)N * DN * sizeof(float), stream);

    k_prep<<<1, 64, 0, stream>>>(small, p.We_, p.ae_, 1.0f / (float)E);
    k_gemm_wmma<<<gbGemm, 128, 0, stream>>>(p.xin, p.W, hbuf, N);
    k_attn_coef<<<gbNH, TB, 0, stream>>>(hbuf, p.as_, p.ad_, asrc, adst, N);
    k_edge_alpha<<<gbEdge, TB, 0, stream>>>(esrc, edst, ea, small, asrc, adst,
                                            etmp, amax, E, EP);
    k_edge_exp<<<gbEdge, TB, 0, stream>>>(edst, etmp, amax, den, E, EP);
    k_edge_aggr<<<gbEH, TB, 0, stream>>>(esrc, edst, etmp, den, hbuf, gout, E, EP);
    k_bn_stats<<<1024, TB, 0, stream>>>(gout, p.b_, small, N);
    k_bn_final<<<1, 64, 0, stream>>>(small, 1.0f / (float)N);
    k_apply<<<gbND, TB, 0, stream>>>(gout, p.b_, small, gamma, beta, p.resid, p.xout, N);
  }
}